// GATNet_69947837383271
// MI455X (gfx1250) — compile-verified
//
#include <hip/hip_runtime.h>
#include <cstdint>
#include <cstddef>

// ---------------- problem constants (match reference) ----------------
#define NN   20000            // nodes
#define NE   320000           // raw edges
#define ET   (NE + NN)        // edges + self loops
#define NF   128              // input features
#define F12  1024             // hidden width (H1*C1)
#define H12  4                // heads, layers 1-2
#define C12  256              // channels/head, layers 1-2
#define H3   6                // heads, layer 3
#define C3   10               // channels/head, layer 3 (= N_CLASS)
#define F3W  (H3 * C3)        // 60
#define NEG_SLOPE 0.2f

typedef float v2f __attribute__((ext_vector_type(2)));
typedef float v8f __attribute__((ext_vector_type(8)));

// ---------------------------------------------------------------------
// Full-precision f32 WMMA GEMM:  C[N,M] = A[N,K] @ B[K,M]   (row-major)
// V_WMMA_F32_16X16X4_F32; one wave computes a 64x16 output tile
// (4 accumulators, shared B fragment). K loop is software-pipelined with
// TWO named fragment sets (ping-pong, prefetch distance 2, no rotation
// copies) so even/odd fragments live in distinct VGPRs and the scheduler
// overlaps each step's load clause with the other step's WMMAs.
// Row/col indices clamped for loads, stores predicated -> any N, any M,
// EXEC all-ones around every WMMA. Requires K % 8 == 0 and K >= 8.
// ---------------------------------------------------------------------
__global__ __launch_bounds__(256) void wmma_gemm_f32(
    const float* __restrict__ A, const float* __restrict__ B,
    float* __restrict__ C, int N, int K, int M)
{
    const int lane = threadIdx.x & 31;
    const int wave = threadIdx.x >> 5;
    const int row0 = blockIdx.x * 64;
    const int col0 = (blockIdx.y * 8 + wave) * 16;
    if (row0 >= N || col0 >= M) return;   // wave-uniform: EXEC stays all-ones

    const int hl = lane >> 4;    // half-wave select (K pair / row+8)
    const int li = lane & 15;    // row (A) / col (B,C)
    const int colB = (col0 + li < M) ? (col0 + li) : (M - 1);   // clamped load col

    // Per-tile A row base pointers, rows clamped so tail tiles read valid memory.
    const float* Ap[4];
#pragma unroll
    for (int t = 0; t < 4; ++t) {
        int r = row0 + t * 16 + li;
        if (r >= N) r = N - 1;
        Ap[t] = A + (size_t)r * K + 2 * hl;
    }
    const float* Bbase = B + colB + (size_t)(2 * hl) * M;  // row 2*hl of k=0

    v8f acc[4];
#pragma unroll
    for (int t = 0; t < 4; ++t) acc[t] = (v8f){0.f,0.f,0.f,0.f,0.f,0.f,0.f,0.f};

    auto loadFrag = [&](int k, v2f (&a)[4], v2f& b) {
        const float* Brow = Bbase + (size_t)k * M;
        b.x = Brow[0];
        b.y = Brow[M];
#pragma unroll
        for (int t = 0; t < 4; ++t) a[t] = *(const v2f*)(Ap[t] + k);
    };
    auto mmaGroup = [&](v2f (&a)[4], v2f& b) {
#pragma unroll
        for (int t = 0; t < 4; ++t)
            acc[t] = __builtin_amdgcn_wmma_f32_16x16x4_f32(
                         false, a[t], false, b, (short)0, acc[t], false, false);
    };

    // ---- pipeline: ping-pong fragment sets, prefetch distance 2 ----
    v2f aE[4], bE, aO[4], bO;
    loadFrag(0, aE, bE);
    loadFrag(4, aO, bO);
    int k = 0;
    for (; k + 8 < K; k += 8) {
        mmaGroup(aE, bE);          // consume step k
        loadFrag(k + 8, aE, bE);   // prefetch step k+8
        mmaGroup(aO, bO);          // consume step k+4
        loadFrag(k + 12, aO, bO);  // prefetch step k+12 (last read: row K-1)
    }
    mmaGroup(aE, bE);              // step K-8
    mmaGroup(aO, bO);              // step K-4

    // D layout: VGPR r -> row = rowTile + r + 8*hl, col = col0 + li
    if (col0 + li < M) {
#pragma unroll
        for (int t = 0; t < 4; ++t) {
            int rowt = row0 + t * 16;
            if (rowt < N) {
                float* Cp = C + (size_t)(rowt + 8 * hl) * M + col0 + li;
#pragma unroll
                for (int r = 0; r < 8; ++r) Cp[(size_t)r * M] = acc[t][r];
            }
        }
    }
}

__global__ void fill_kernel(float* __restrict__ p, float v, long long n)
{
    long long i = (long long)blockIdx.x * blockDim.x + threadIdx.x;
    if (i < n) p[i] = v;
}

// al_s[n,h] = sum_c h[n,h,c]*a_src[h,c]; al_d likewise. One wave per (n,h).
__global__ void alpha_kernel(const float* __restrict__ hfeat,
                             const float* __restrict__ a_src,
                             const float* __restrict__ a_dst,
                             float* __restrict__ al_s, float* __restrict__ al_d,
                             int Nn, int H, int C)
{
    int wid  = blockIdx.x * (blockDim.x >> 5) + (threadIdx.x >> 5);
    int lane = threadIdx.x & 31;
    if (wid >= Nn * H) return;           // wave-uniform
    int n = wid / H, h = wid % H;
    const float* hp = hfeat + (size_t)n * H * C + (size_t)h * C;
    float s = 0.f, d = 0.f;
    for (int c = lane; c < C; c += 32) {
        float v = hp[c];
        s += v * a_src[h * C + c];
        d += v * a_dst[h * C + c];
    }
#pragma unroll
    for (int off = 16; off > 0; off >>= 1) {
        s += __shfl_down(s, off, 32);
        d += __shfl_down(d, off, 32);
    }
    if (lane == 0) { al_s[wid] = s; al_d[wid] = d; }
}

__device__ inline void atomicMaxF32(float* addr, float val)
{
    unsigned int* ua = (unsigned int*)addr;
    unsigned int old = *ua;
    while (__uint_as_float(old) < val) {
        unsigned int assumed = old;
        old = atomicCAS(ua, assumed, __float_as_uint(val));
        if (old == assumed) break;
    }
}

// e[i,h] = leaky_relu(al_s[src]+al_d[dst]); m[dst,h] = max(...)
__global__ void edge_score(const int* __restrict__ src, const int* __restrict__ dst,
                           const float* __restrict__ al_s, const float* __restrict__ al_d,
                           float* __restrict__ e, float* __restrict__ mbuf,
                           int E, int Nn, int H)
{
    int i = blockIdx.x * blockDim.x + threadIdx.x;
    if (i >= E + Nn) return;
    int s, d;
    if (i < E) { s = src[i]; d = dst[i]; } else { s = d = i - E; }
    for (int h = 0; h < H; ++h) {
        float v = al_s[s * H + h] + al_d[d * H + h];
        v = (v > 0.f) ? v : NEG_SLOPE * v;
        e[(size_t)i * H + h] = v;
        atomicMaxF32(&mbuf[d * H + h], v);
    }
}

// e[i,h] <- exp(e - m[dst]);  den[dst,h] += e   (in-place)
__global__ void edge_exp(const int* __restrict__ dst, const float* __restrict__ mbuf,
                         float* __restrict__ e, float* __restrict__ den,
                         int E, int Nn, int H)
{
    int i = blockIdx.x * blockDim.x + threadIdx.x;
    if (i >= E + Nn) return;
    int d = (i < E) ? dst[i] : (i - E);
    for (int h = 0; h < H; ++h) {
        float ex = __expf(e[(size_t)i * H + h] - mbuf[d * H + h]);
        e[(size_t)i * H + h] = ex;
        atomicAdd(&den[d * H + h], ex);
    }
}

// agg[dst, h, c] += h[src, h, c] * (ex/(den+eps)) * ew.  One wave per edge.
__global__ void edge_aggregate(const int* __restrict__ src, const int* __restrict__ dst,
                               const float* __restrict__ ew,
                               const float* __restrict__ ex, const float* __restrict__ den,
                               const float* __restrict__ hfeat, float* __restrict__ agg,
                               int E, int Nn, int H, int C)
{
    int wid  = blockIdx.x * (blockDim.x >> 5) + (threadIdx.x >> 5);
    int lane = threadIdx.x & 31;
    if (wid >= E + Nn) return;           // wave-uniform
    int s, d; float w;
    if (wid < E) { s = src[wid]; d = dst[wid]; w = ew[wid]; }
    else         { s = d = wid - E; w = 1.f; }
    const int HC = H * C;
    const float* hp = hfeat + (size_t)s * HC;
    float*       ap = agg   + (size_t)d * HC;
    for (int j = lane; j < HC; j += 32) {
        int h = j / C;
        float coef = ex[(size_t)wid * H + h] / (den[d * H + h] + 1e-16f) * w;
        atomicAdd(&ap[j], hp[j] * coef);
    }
}

// out = elu(agg + b + skip + lb)   (layers 1-2)
__global__ void finalize_mid(const float* __restrict__ agg, const float* __restrict__ b,
                             const float* __restrict__ lb, const float* __restrict__ skip,
                             float* __restrict__ out, int Nn, int F)
{
    int idx = blockIdx.x * blockDim.x + threadIdx.x;
    if (idx >= Nn * F) return;
    int j = idx % F;
    float v = agg[idx] + b[j] + skip[idx] + lb[j];
    out[idx] = (v > 0.f) ? v : (expf(v) - 1.f);
}

// out = mean_h(agg) + b + skip + lb   (layer 3)
__global__ void finalize_out(const float* __restrict__ agg, const float* __restrict__ b,
                             const float* __restrict__ lb, const float* __restrict__ skip,
                             float* __restrict__ out, int Nn, int H, int C)
{
    int idx = blockIdx.x * blockDim.x + threadIdx.x;
    if (idx >= Nn * C) return;
    int n = idx / C, c = idx % C;
    float s = 0.f;
    for (int h = 0; h < H; ++h) s += agg[(size_t)n * H * C + h * C + c];
    out[idx] = s * (1.f / H) + b[c] + lb[c] + skip[idx];
}

// ---------------------------------------------------------------------
extern "C" void kernel_launch(void* const* d_in, const int* in_sizes, int n_in,
                              void* d_out, int out_size, void* d_ws, size_t ws_size,
                              hipStream_t stream)
{
    (void)in_sizes; (void)n_in; (void)out_size; (void)ws_size;
    const float* x   = (const float*)d_in[0];
    const int*   ei  = (const int*)  d_in[1];
    const float* ew  = (const float*)d_in[2];
    const float* W1  = (const float*)d_in[3];
    const float* a1s = (const float*)d_in[4];
    const float* a1d = (const float*)d_in[5];
    const float* b1  = (const float*)d_in[6];
    const float* lw1 = (const float*)d_in[7];
    const float* lb1 = (const float*)d_in[8];
    const float* W2  = (const float*)d_in[9];
    const float* a2s = (const float*)d_in[10];
    const float* a2d = (const float*)d_in[11];
    const float* b2  = (const float*)d_in[12];
    const float* lw2 = (const float*)d_in[13];
    const float* lb2 = (const float*)d_in[14];
    const float* W3  = (const float*)d_in[15];
    const float* a3s = (const float*)d_in[16];
    const float* a3d = (const float*)d_in[17];
    const float* b3  = (const float*)d_in[18];
    const float* lw3 = (const float*)d_in[19];
    const float* lb3 = (const float*)d_in[20];
    const int* src = ei;
    const int* dst = ei + NE;
    float* out = (float*)d_out;

    // workspace carve-up (floats)
    float* ws = (float*)d_ws;
    size_t o = 0;
    auto take = [&](size_t nf) { float* p = ws + o; o += (nf + 63) & ~(size_t)63; return p; };
    float* hw    = take((size_t)NN * F12);   // x@W   (per layer)
    float* skip  = take((size_t)NN * F12);   // x@lw  (per layer)
    float* agg   = take((size_t)NN * F12);   // segment-sum accumulator
    float* hcur  = take((size_t)NN * F12);   // layer output features
    float* al_s  = take((size_t)NN * H3);
    float* al_d  = take((size_t)NN * H3);
    float* mbuf  = take((size_t)NN * H3);
    float* den   = take((size_t)NN * H3);
    float* ebuf  = take((size_t)ET * H3);    // e then ex (in place)
    float* hw3   = take((size_t)NN * F3W);
    float* skip3 = take((size_t)NN * C3);
    float* agg3  = take((size_t)NN * F3W);

    auto cdiv = [](long long a, long long b) { return (int)((a + b - 1) / b); };
    const int TB = 256;
    dim3 blk(TB);
    const int gemmRows = cdiv(NN, 64);       // 64-row tiles per wave

    // --------------- mid layer runner (H=4, C=256, F=1024) ---------------
    auto run_mid = [&](const float* xin, int K,
                       const float* W, const float* lw,
                       const float* as_, const float* ad_,
                       const float* b, const float* lb, float* hout)
    {
        dim3 gg(gemmRows, F12 / (16 * 8));
        wmma_gemm_f32<<<gg, blk, 0, stream>>>(xin, W,  hw,   NN, K, F12);
        wmma_gemm_f32<<<gg, blk, 0, stream>>>(xin, lw, skip, NN, K, F12);

        alpha_kernel<<<cdiv((long long)NN * H12, 8), blk, 0, stream>>>(
            hw, as_, ad_, al_s, al_d, NN, H12, C12);

        fill_kernel<<<cdiv((long long)NN * H12, TB), blk, 0, stream>>>(mbuf, -3.0e38f, (long long)NN * H12);
        fill_kernel<<<cdiv((long long)NN * H12, TB), blk, 0, stream>>>(den, 0.f, (long long)NN * H12);
        fill_kernel<<<cdiv((long long)NN * F12, TB), blk, 0, stream>>>(agg, 0.f, (long long)NN * F12);

        edge_score<<<cdiv(ET, TB), blk, 0, stream>>>(src, dst, al_s, al_d, ebuf, mbuf, NE, NN, H12);
        edge_exp  <<<cdiv(ET, TB), blk, 0, stream>>>(dst, mbuf, ebuf, den, NE, NN, H12);
        edge_aggregate<<<cdiv(ET, 8), blk, 0, stream>>>(src, dst, ew, ebuf, den, hw, agg,
                                                        NE, NN, H12, C12);

        finalize_mid<<<cdiv((long long)NN * F12, TB), blk, 0, stream>>>(
            agg, b, lb, skip, hout, NN, F12);
    };

    // layer 1: x[NN,128] -> hcur[NN,1024]
    run_mid(x, NF, W1, lw1, a1s, a1d, b1, lb1, hcur);
    // layer 2: hcur -> hcur (GEMMs consume hcur before finalize overwrites it)
    run_mid(hcur, F12, W2, lw2, a2s, a2d, b2, lb2, hcur);

    // --------------- layer 3 (H=6, C=10, mean over heads) ---------------
    {
        dim3 g3(gemmRows, 1);    // M=60 -> 4 col tiles (waves 4..7 idle-uniform)
        wmma_gemm_f32<<<g3, blk, 0, stream>>>(hcur, W3,  hw3,   NN, F12, F3W);
        wmma_gemm_f32<<<g3, blk, 0, stream>>>(hcur, lw3, skip3, NN, F12, C3);
    }

    alpha_kernel<<<cdiv((long long)NN * H3, 8), blk, 0, stream>>>(
        hw3, a3s, a3d, al_s, al_d, NN, H3, C3);

    fill_kernel<<<cdiv((long long)NN * H3, TB), blk, 0, stream>>>(mbuf, -3.0e38f, (long long)NN * H3);
    fill_kernel<<<cdiv((long long)NN * H3, TB), blk, 0, stream>>>(den, 0.f, (long long)NN * H3);
    fill_kernel<<<cdiv((long long)NN * F3W, TB), blk, 0, stream>>>(agg3, 0.f, (long long)NN * F3W);

    edge_score<<<cdiv(ET, TB), blk, 0, stream>>>(src, dst, al_s, al_d, ebuf, mbuf, NE, NN, H3);
    edge_exp  <<<cdiv(ET, TB), blk, 0, stream>>>(dst, mbuf, ebuf, den, NE, NN, H3);
    edge_aggregate<<<cdiv(ET, 8), blk, 0, stream>>>(src, dst, ew, ebuf, den, hw3, agg3,
                                                    NE, NN, H3, C3);

    finalize_out<<<cdiv((long long)NN * C3, TB), blk, 0, stream>>>(
        agg3, b3, lb3, skip3, out, NN, H3, C3);
}